// LSTM_FOL_41266045780254
// MI455X (gfx1250) — compile-verified
//
#include <hip/hip_runtime.h>
#include <math.h>

// ---------------- problem constants ----------------
#define NB    256      // batch
#define NT    128      // sequence length
#define NE    300      // embedding dim
#define NH    300      // hidden
#define N4H   1200     // 4*H (gates)
#define NP    1216     // 4H padded to 76*16 for WMMA tiles
#define NH2   600      // 2*H
#define NH2P  608      // 2H padded to 38*16
#define NTREE 22
#define HPAD  304      // padded h/c row stride (even, 16B-friendly)

typedef __attribute__((ext_vector_type(2))) float v2f;
typedef __attribute__((ext_vector_type(8))) float v8f;

// D(16x16,f32) = A(16x4,f32) * B(4x16,f32) + C  — CDNA5 f32 WMMA
__device__ __forceinline__ v8f wmma4(v2f a, v2f b, v8f c) {
  return __builtin_amdgcn_wmma_f32_16x16x4_f32(
      /*neg_a=*/false, a, /*neg_b=*/false, b,
      /*c_mod=*/(short)0, c, /*reuse_a=*/false, /*reuse_b=*/false);
}

__device__ __forceinline__ float sigf(float x)  { return 1.0f / (1.0f + expf(-x)); }
__device__ __forceinline__ float leakyf(float v){ return v >= 0.0f ? v : -0.2f * v; }

// h_t[b][j] = concat(fwd_out[b,t,:], bwd_out[b,t,:])[j],  j in [0,600)
__device__ __forceinline__ float htsel(const float* __restrict__ fwd_out,
                                       const float* __restrict__ bwd_out,
                                       int b, int j, int t) {
  return (j < NH) ? fwd_out[((size_t)b * NTREE + t) * NH + j]
                  : bwd_out[((size_t)b * NTREE + t) * NH + (j - NH)];
}

// A-fragment (2 consecutive K, even-aligned) of concat(state, h_t)
__device__ __forceinline__ v2f xcat2(const float* __restrict__ state,
                                     const float* __restrict__ fwd_out,
                                     const float* __restrict__ bwd_out,
                                     int row, int ka, int t) {
  const float* p;
  if (ka < NH2) {
    p = state + (size_t)row * NH2P + ka;
  } else {
    int j = ka - NH2;                       // ka even: pair never straddles a region
    p = (j < NH) ? fwd_out + ((size_t)row * NTREE + t) * NH + j
                 : bwd_out + ((size_t)row * NTREE + t) * NH + (j - NH);
  }
  v2f a; a.x = p[0]; a.y = p[1]; return a;
}

// ---------------- utility kernels ----------------
__global__ void k_zero(float* __restrict__ p, int n) {
  int i = blockIdx.x * blockDim.x + threadIdx.x;
  if (i < n) p[i] = 0.0f;
}

__global__ void k_lens(const int* __restrict__ idx, int* __restrict__ lens) {
  int b = threadIdx.x;
  if (b < NB) {
    int c = 0;
    for (int t = 0; t < NT; ++t) c += (idx[b * NT + t] != 0);
    lens[b] = c;
  }
}

// K-pair-interleaved transpose: element (k,n) of W^T stored at
// dst[((k>>1)*NPd + n)*2 + (k&1)]  -> one b64 load fetches K-pair per lane.
__global__ void k_transpose(const float* __restrict__ src, float* __restrict__ dst,
                            int N, int K, int NPd) {
  int tid = blockIdx.x * blockDim.x + threadIdx.x;
  if (tid >= K * NPd) return;
  int k = tid / NPd, n = tid % NPd;
  float v = (n < N) ? src[(size_t)n * K + k] : 0.0f;
  dst[((size_t)(k >> 1) * NPd + n) * 2 + (k & 1)] = v;
}

__global__ void k_bias(const float* __restrict__ a, const float* __restrict__ b,
                       float* __restrict__ o) {
  int n = blockIdx.x * blockDim.x + threadIdx.x;
  if (n < NP) o[n] = (n < N4H) ? (a[n] + b[n]) : 0.0f;
}

// ---------------- LSTM per-step gate GEMM ----------------
// g[b,n] = bsum[n] + x_t[b,:]·WihT[:,n] + h[b,:]·WhhT[:,n]
// one wave -> 32 rows x 64 cols (8 accumulators), weights in K-pair layout.
__global__ void k_step_gemm(const int* __restrict__ idx, const float* __restrict__ embed,
                            const int* __restrict__ lens,
                            const float* __restrict__ WihT_f, const float* __restrict__ WhhT_f,
                            const float* __restrict__ WihT_b, const float* __restrict__ WhhT_b,
                            const float* __restrict__ bsum_f, const float* __restrict__ bsum_b,
                            const float* __restrict__ h_f, const float* __restrict__ h_b,
                            float* __restrict__ g_f, float* __restrict__ g_b,
                            int t, int doFwd) {
  const int wavesPerDir = 8 * 19;             // 8 M-tiles(32 rows) x 19 N-groups(64 cols)
  int wave  = (blockIdx.x * blockDim.x + threadIdx.x) >> 5;
  int total = (doFwd ? 2 : 1) * wavesPerDir;
  if (wave >= total) return;
  int dir = doFwd ? (wave / wavesPerDir) : 1;
  int wi  = doFwd ? (wave % wavesPerDir) : wave;
  int mBase = (wi / 19) << 5;
  int nBase = (wi % 19) << 6;
  int lane = threadIdx.x & 31, half = lane >> 4, l16 = lane & 15;
  int k0   = 2 * half;
  int row0 = mBase + l16, row1 = row0 + 16;

  int pos0, pos1;
  if (dir == 0) {
    pos0 = t; pos1 = t;
  } else {
    int L0 = lens[row0], L1 = lens[row1];
    pos0 = L0 - 1 - t; pos0 = pos0 < 0 ? 0 : (pos0 > NT - 1 ? NT - 1 : pos0);
    pos1 = L1 - 1 - t; pos1 = pos1 < 0 ? 0 : (pos1 > NT - 1 ? NT - 1 : pos1);
  }
  const float* x0 = embed + (size_t)idx[row0 * NT + pos0] * NE;
  const float* x1 = embed + (size_t)idx[row1 * NT + pos1] * NE;
  const float* hb = (dir ? h_b : h_f);
  const float* h0 = hb + (size_t)row0 * HPAD;
  const float* h1 = hb + (size_t)row1 * HPAD;
  const float* WiT = dir ? WihT_b : WihT_f;
  const float* WhT = dir ? WhhT_b : WhhT_f;
  const float* bs  = dir ? bsum_b : bsum_f;
  float*       g   = dir ? g_b    : g_f;

  v8f a00 = {}, a01 = {}, a02 = {}, a03 = {};
  v8f a10 = {}, a11 = {}, a12 = {}, a13 = {};

  for (int kb = 0; kb < NE; kb += 4) {               // x @ WihT
    v2f aL = *(const v2f*)(x0 + kb + k0);
    v2f aH = *(const v2f*)(x1 + kb + k0);
    const float* w = WiT + ((size_t)(kb / 2 + half) * NP + nBase + l16) * 2;
    v2f b0 = *(const v2f*)(w);
    v2f b1 = *(const v2f*)(w + 32);
    v2f b2 = *(const v2f*)(w + 64);
    v2f b3 = *(const v2f*)(w + 96);
    a00 = wmma4(aL, b0, a00); a01 = wmma4(aL, b1, a01);
    a02 = wmma4(aL, b2, a02); a03 = wmma4(aL, b3, a03);
    a10 = wmma4(aH, b0, a10); a11 = wmma4(aH, b1, a11);
    a12 = wmma4(aH, b2, a12); a13 = wmma4(aH, b3, a13);
  }
  for (int kb = 0; kb < NH; kb += 4) {               // h @ WhhT
    v2f aL = *(const v2f*)(h0 + kb + k0);
    v2f aH = *(const v2f*)(h1 + kb + k0);
    const float* w = WhT + ((size_t)(kb / 2 + half) * NP + nBase + l16) * 2;
    v2f b0 = *(const v2f*)(w);
    v2f b1 = *(const v2f*)(w + 32);
    v2f b2 = *(const v2f*)(w + 64);
    v2f b3 = *(const v2f*)(w + 96);
    a00 = wmma4(aL, b0, a00); a01 = wmma4(aL, b1, a01);
    a02 = wmma4(aL, b2, a02); a03 = wmma4(aL, b3, a03);
    a10 = wmma4(aH, b0, a10); a11 = wmma4(aH, b1, a11);
    a12 = wmma4(aH, b2, a12); a13 = wmma4(aH, b3, a13);
  }

  float bi0 = bs[nBase + l16];
  float bi1 = bs[nBase + l16 + 16];
  float bi2 = bs[nBase + l16 + 32];
  float bi3 = bs[nBase + l16 + 48];
#pragma unroll
  for (int j = 0; j < 8; ++j) {
    int r0 = mBase + j + 8 * half;                   // D layout: VGPR j -> M=j (+8 hi half)
    float* g0 = g + (size_t)r0 * NP + nBase + l16;
    g0[0]  = a00[j] + bi0;
    g0[16] = a01[j] + bi1;
    g0[32] = a02[j] + bi2;
    g0[48] = a03[j] + bi3;
    float* g1 = g0 + (size_t)16 * NP;
    g1[0]  = a10[j] + bi0;
    g1[16] = a11[j] + bi1;
    g1[32] = a12[j] + bi2;
    g1[48] = a13[j] + bi3;
  }
}

// ---------------- LSTM pointwise gate update ----------------
__global__ void k_step_update(const int* __restrict__ lens,
                              const float* __restrict__ g_f, const float* __restrict__ g_b,
                              float* __restrict__ h_f, float* __restrict__ c_f,
                              float* __restrict__ h_b, float* __restrict__ c_b,
                              float* __restrict__ fwd_out, float* __restrict__ bwd_out,
                              int t, int doFwd) {
  const int per = NB * NH;
  int tid = blockIdx.x * blockDim.x + threadIdx.x;
  int total = (doFwd ? 2 : 1) * per;
  if (tid >= total) return;
  int dir = doFwd ? (tid / per) : 1;
  int s   = doFwd ? (tid % per) : tid;
  int b = s / NH, e = s % NH;
  int L = lens[b];
  if (t >= L) return;                                 // masked step
  const float* g = (dir ? g_b : g_f) + (size_t)b * NP;
  float iv = g[e], fv = g[NH + e], gv = g[2 * NH + e], ov = g[3 * NH + e];
  float* cp = (dir ? c_b : c_f) + (size_t)b * HPAD + e;
  float* hp = (dir ? h_b : h_f) + (size_t)b * HPAD + e;
  float cn = sigf(fv) * (*cp) + sigf(iv) * tanhf(gv);
  float hn = sigf(ov) * tanhf(cn);
  *cp = cn;
  *hp = hn;
  if (dir == 0) {
    if (t < NTREE) fwd_out[((size_t)b * NTREE + t) * NH + e] = hn;
  } else {
    int tp = L - 1 - t;                               // reversed position
    if (tp < NTREE) bwd_out[((size_t)b * NTREE + tp) * NH + e] = hn;
  }
}

// ---------------- tree phase ----------------
__global__ void k_state_init(const float* __restrict__ fwd_out,
                             const float* __restrict__ bwd_out,
                             float* __restrict__ state) {
  int tid = blockIdx.x * blockDim.x + threadIdx.x;
  if (tid >= NB * NH2P) return;
  int b = tid / NH2P, j = tid % NH2P;
  float v = 0.0f;
  if (j < NH)       v = fwd_out[(size_t)b * NTREE * NH + j];
  else if (j < NH2) v = bwd_out[(size_t)b * NTREE * NH + (j - NH)];
  state[tid] = v;
}

// wa_s[b] = state[b]·and_W + and_b ; wa_h[b] = h_t[b]·and_W + and_b  (one wave per row)
__global__ void k_tree_dots(const float* __restrict__ state,
                            const float* __restrict__ fwd_out, const float* __restrict__ bwd_out,
                            const float* __restrict__ and_W, const float* __restrict__ and_b,
                            float* __restrict__ wa_s, float* __restrict__ wa_h, int t) {
  int wave = (blockIdx.x * blockDim.x + threadIdx.x) >> 5;
  if (wave >= NB) return;
  int lane = threadIdx.x & 31;
  float ss = 0.0f, sh = 0.0f;
  for (int j = lane; j < NH2; j += 32) {
    float w = and_W[j];
    ss += w * state[(size_t)wave * NH2P + j];
    sh += w * htsel(fwd_out, bwd_out, wave, j, t);
  }
  for (int m = 16; m >= 1; m >>= 1) {
    ss += __shfl_xor(ss, m);
    sh += __shfl_xor(sh, m);
  }
  if (lane == 0) {
    wa_s[wave] = ss + and_b[0];
    wa_h[wave] = sh + and_b[0];
  }
}

// or_res[b,n] = concat(state,h_t)[b,:] @ orWT[:,n] + or_b[n]   (K=1200, 32x32 tiles)
__global__ void k_tree_or(const float* __restrict__ state,
                          const float* __restrict__ fwd_out, const float* __restrict__ bwd_out,
                          const float* __restrict__ orWT, const float* __restrict__ or_b,
                          float* __restrict__ or_res, int t) {
  const int waves = 8 * 19;
  int wave = (blockIdx.x * blockDim.x + threadIdx.x) >> 5;
  if (wave >= waves) return;
  int mBase = (wave / 19) << 5;
  int nBase = (wave % 19) << 5;
  int lane = threadIdx.x & 31, half = lane >> 4, l16 = lane & 15;
  int k0 = 2 * half;
  int row0 = mBase + l16, row1 = row0 + 16;

  v8f a00 = {}, a01 = {}, a10 = {}, a11 = {};
  for (int kb = 0; kb < N4H; kb += 4) {
    int ka = kb + k0;
    v2f aL = xcat2(state, fwd_out, bwd_out, row0, ka, t);
    v2f aH = xcat2(state, fwd_out, bwd_out, row1, ka, t);
    const float* w = orWT + ((size_t)(kb / 2 + half) * NH2P + nBase + l16) * 2;
    v2f b0 = *(const v2f*)(w);
    v2f b1 = *(const v2f*)(w + 32);
    a00 = wmma4(aL, b0, a00); a01 = wmma4(aL, b1, a01);
    a10 = wmma4(aH, b0, a10); a11 = wmma4(aH, b1, a11);
  }
  int col0 = nBase + l16;
  float bi0 = (col0      < NH2) ? or_b[col0]      : 0.0f;
  float bi1 = (col0 + 16 < NH2) ? or_b[col0 + 16] : 0.0f;
#pragma unroll
  for (int j = 0; j < 8; ++j) {
    int r0 = mBase + j + 8 * half;
    or_res[(size_t)r0 * NH2P + col0]        = a00[j] + bi0;
    or_res[(size_t)r0 * NH2P + col0 + 16]   = a01[j] + bi1;
    or_res[((size_t)r0 + 16) * NH2P + col0]      = a10[j] + bi0;
    or_res[((size_t)r0 + 16) * NH2P + col0 + 16] = a11[j] + bi1;
  }
}

// not_res = leaky(or_res) @ notWT + not_b, then fused and/or/not select into state
__global__ void k_tree_not(const float* __restrict__ or_res,
                           const float* __restrict__ notWT, const float* __restrict__ not_b,
                           const float* __restrict__ fwd_out, const float* __restrict__ bwd_out,
                           const float* __restrict__ wa_s, const float* __restrict__ wa_h,
                           const int* __restrict__ idx, float* __restrict__ state, int t) {
  const int waves = 8 * 19;
  int wave = (blockIdx.x * blockDim.x + threadIdx.x) >> 5;
  if (wave >= waves) return;
  int mBase = (wave / 19) << 5;
  int nBase = (wave % 19) << 5;
  int lane = threadIdx.x & 31, half = lane >> 4, l16 = lane & 15;
  int k0 = 2 * half;
  int row0 = mBase + l16, row1 = row0 + 16;

  v8f a00 = {}, a01 = {}, a10 = {}, a11 = {};
  for (int kb = 0; kb < NH2; kb += 4) {
    int ka = kb + k0;
    v2f aL = *(const v2f*)(or_res + (size_t)row0 * NH2P + ka);
    v2f aH = *(const v2f*)(or_res + (size_t)row1 * NH2P + ka);
    aL.x = leakyf(aL.x); aL.y = leakyf(aL.y);
    aH.x = leakyf(aH.x); aH.y = leakyf(aH.y);
    const float* w = notWT + ((size_t)(kb / 2 + half) * NH2P + nBase + l16) * 2;
    v2f b0 = *(const v2f*)(w);
    v2f b1 = *(const v2f*)(w + 32);
    a00 = wmma4(aL, b0, a00); a01 = wmma4(aL, b1, a01);
    a10 = wmma4(aH, b0, a10); a11 = wmma4(aH, b1, a11);
  }
  int col0 = nBase + l16;
  int col1 = col0 + 16;
  float nb0 = (col0 < NH2) ? not_b[col0] : 0.0f;
  float nb1 = (col1 < NH2) ? not_b[col1] : 0.0f;
#pragma unroll
  for (int sub = 0; sub < 2; ++sub) {
#pragma unroll
    for (int j = 0; j < 8; ++j) {
      int r = mBase + 16 * sub + j + 8 * half;
      int tok = idx[r * NT + t];
      int m = tok % 3;
      float was = wa_s[r], wah = wa_h[r];
      float n0 = (sub ? a10[j] : a00[j]) + nb0;
      float n1 = (sub ? a11[j] : a01[j]) + nb1;
      if (col0 < NH2) {
        float st = state[(size_t)r * NH2P + col0];
        float hv = htsel(fwd_out, bwd_out, r, col0, t);
        float andv = was * st + wah * hv;
        float orv  = or_res[(size_t)r * NH2P + col0];
        float nv = (m == 0) ? andv : ((m == 1) ? orv : n0);
        state[(size_t)r * NH2P + col0] = (tok == 0) ? st : nv;
      }
      if (col1 < NH2) {
        float st = state[(size_t)r * NH2P + col1];
        float hv = htsel(fwd_out, bwd_out, r, col1, t);
        float andv = was * st + wah * hv;
        float orv  = or_res[(size_t)r * NH2P + col1];
        float nv = (m == 0) ? andv : ((m == 1) ? orv : n1);
        state[(size_t)r * NH2P + col1] = (tok == 0) ? st : nv;
      }
    }
  }
}

__global__ void k_dense(const float* __restrict__ state,
                        const float* __restrict__ dW, const float* __restrict__ db,
                        float* __restrict__ out) {
  int tid = blockIdx.x * blockDim.x + threadIdx.x;
  if (tid >= NB * 3) return;
  int b = tid / 3, p = tid % 3;
  float s = db[p];
  const float* sr = state + (size_t)b * NH2P;
  const float* wr = dW + (size_t)p * NH2;
  for (int j = 0; j < NH2; ++j) s += sr[j] * wr[j];
  out[tid] = s;
}

// ---------------- host orchestration ----------------
extern "C" void kernel_launch(void* const* d_in, const int* in_sizes, int n_in,
                              void* d_out, int out_size, void* d_ws, size_t ws_size,
                              hipStream_t stream) {
  (void)in_sizes; (void)n_in; (void)out_size; (void)ws_size;
  const int*   idx     = (const int*)  d_in[0];
  const float* embed   = (const float*)d_in[1];
  const float* Wih_f   = (const float*)d_in[2];
  const float* Whh_f   = (const float*)d_in[3];
  const float* bih_f   = (const float*)d_in[4];
  const float* bhh_f   = (const float*)d_in[5];
  const float* Wih_b   = (const float*)d_in[6];
  const float* Whh_b   = (const float*)d_in[7];
  const float* bih_b   = (const float*)d_in[8];
  const float* bhh_b   = (const float*)d_in[9];
  const float* not_W   = (const float*)d_in[10];
  const float* not_b   = (const float*)d_in[11];
  const float* and_W   = (const float*)d_in[12];
  const float* and_b   = (const float*)d_in[13];
  const float* or_W    = (const float*)d_in[14];
  const float* or_b    = (const float*)d_in[15];
  const float* dense_W = (const float*)d_in[16];
  const float* dense_b = (const float*)d_in[17];
  float* out = (float*)d_out;

  // ---- carve workspace (region sizes are multiples of 64 floats) ----
  float* base = (float*)d_ws;
  size_t off = 0;
  auto take = [&](size_t n) { float* p = base + off; off += n; return p; };
  float* WihT_f  = take((size_t)NE  * NP);     // K-pair-interleaved 300x1216
  float* WhhT_f  = take((size_t)NH  * NP);
  float* WihT_b  = take((size_t)NE  * NP);
  float* WhhT_b  = take((size_t)NH  * NP);
  float* bsum_f  = take(NP);
  float* bsum_b  = take(NP);
  float* orWT    = take((size_t)N4H * NH2P);   // 1200x608 (paired)
  float* notWT   = take((size_t)NH2 * NH2P);   // 600x608 (paired)
  float* g_f     = take((size_t)NB * NP);
  float* g_b     = take((size_t)NB * NP);
  float* zeroReg = base + off;                 // contiguous zero-init region
  float* h_f     = take((size_t)NB * HPAD);
  float* c_f     = take((size_t)NB * HPAD);
  float* h_b     = take((size_t)NB * HPAD);
  float* c_b     = take((size_t)NB * HPAD);
  float* fwd_out = take((size_t)NB * NTREE * NH);
  float* bwd_out = take((size_t)NB * NTREE * NH);
  size_t zeroN   = (size_t)(base + off - zeroReg);
  float* state   = take((size_t)NB * NH2P);
  float* or_res  = take((size_t)NB * NH2P);
  float* wa_s    = take(256);
  float* wa_h    = take(256);
  int*   lens    = (int*)take(256);

  const dim3 blk(256);

  // prologue
  k_zero<<<(int)((zeroN + 255) / 256), blk, 0, stream>>>(zeroReg, (int)zeroN);
  k_lens<<<1, blk, 0, stream>>>(idx, lens);
  k_transpose<<<((NE  * NP   + 255) / 256), blk, 0, stream>>>(Wih_f, WihT_f, N4H, NE,  NP);
  k_transpose<<<((NH  * NP   + 255) / 256), blk, 0, stream>>>(Whh_f, WhhT_f, N4H, NH,  NP);
  k_transpose<<<((NE  * NP   + 255) / 256), blk, 0, stream>>>(Wih_b, WihT_b, N4H, NE,  NP);
  k_transpose<<<((NH  * NP   + 255) / 256), blk, 0, stream>>>(Whh_b, WhhT_b, N4H, NH,  NP);
  k_transpose<<<((N4H * NH2P + 255) / 256), blk, 0, stream>>>(or_W,  orWT,  NH2, N4H, NH2P);
  k_transpose<<<((NH2 * NH2P + 255) / 256), blk, 0, stream>>>(not_W, notWT, NH2, NH2, NH2P);
  k_bias<<<(NP + 255) / 256, blk, 0, stream>>>(bih_f, bhh_f, bsum_f);
  k_bias<<<(NP + 255) / 256, blk, 0, stream>>>(bih_b, bhh_b, bsum_b);

  // recurrent phase: fwd only needs t<22 (only out_states[:, :22] is consumed);
  // bwd runs all 128 steps, storing h only for reversed positions < 22.
  for (int t = 0; t < NT; ++t) {
    int doF = (t < NTREE) ? 1 : 0;
    int waves = (doF ? 2 : 1) * 8 * 19;
    k_step_gemm<<<(waves + 7) / 8, blk, 0, stream>>>(
        idx, embed, lens, WihT_f, WhhT_f, WihT_b, WhhT_b,
        bsum_f, bsum_b, h_f, h_b, g_f, g_b, t, doF);
    int thr = (doF ? 2 : 1) * NB * NH;
    k_step_update<<<(thr + 255) / 256, blk, 0, stream>>>(
        lens, g_f, g_b, h_f, c_f, h_b, c_b, fwd_out, bwd_out, t, doF);
  }

  // tree phase
  k_state_init<<<(NB * NH2P + 255) / 256, blk, 0, stream>>>(fwd_out, bwd_out, state);
  for (int t = 1; t < NTREE; ++t) {
    k_tree_dots<<<(NB * 32 + 255) / 256, blk, 0, stream>>>(
        state, fwd_out, bwd_out, and_W, and_b, wa_s, wa_h, t);
    k_tree_or<<<(8 * 19 * 32) / 256, blk, 0, stream>>>(
        state, fwd_out, bwd_out, orWT, or_b, or_res, t);
    k_tree_not<<<(8 * 19 * 32) / 256, blk, 0, stream>>>(
        or_res, notWT, not_b, fwd_out, bwd_out, wa_s, wa_h, idx, state, t);
  }

  k_dense<<<(NB * 3 + 255) / 256, blk, 0, stream>>>(state, dense_W, dense_b, out);
}